// HyperbolicGraphPooling_56573309223549
// MI455X (gfx1250) — compile-verified
//
#include <hip/hip_runtime.h>

#define CHANNELS        256
#define NUM_GRAPHS      2048
#define WAVE_SIZE       32
#define WAVES_PER_BLOCK 4
#define BLOCK_THREADS   (WAVES_PER_BLOCK * WAVE_SIZE)
#define ROWS_PER_WAVE   128
#define GROUP_ROWS      4
#define GROUP_FLOATS    (GROUP_ROWS * CHANNELS)          /* 1024 floats = 4KB  */
#define WAVE_LDS_FLOATS (2 * GROUP_FLOATS)               /* 2048 floats = 8KB  */
#define BLOCK_LDS_FLOATS (WAVES_PER_BLOCK * WAVE_LDS_FLOATS) /* 32KB per block */

// ---------------------------------------------------------------------------
// CDNA5 async global->LDS copy (ASYNCcnt-tracked) + counter waits.
// Builtin signature (from hipcc diagnostic): (v4i as1*, v4i as3*, Imm, Imm).
// ---------------------------------------------------------------------------
template <int OFF>
__device__ __forceinline__ void async_load_b128(const float* gsrc,
                                                float* ldst) {
#if defined(__AMDGCN__)
#if __has_builtin(__builtin_amdgcn_global_load_async_to_lds_b128)
  typedef int v4i __attribute__((vector_size(16)));
  __builtin_amdgcn_global_load_async_to_lds_b128(
      (__attribute__((address_space(1))) v4i*)gsrc,
      (__attribute__((address_space(3))) v4i*)ldst,
      OFF, 0);
#else
  unsigned lds_off =
      (unsigned)(unsigned long)(__attribute__((address_space(3))) char*)ldst;
  unsigned long long gaddr = (unsigned long long)gsrc;
  asm volatile("global_load_async_to_lds_b128 %0, %1, off offset:%2"
               :
               : "v"(lds_off), "v"(gaddr), "i"(OFF)
               : "memory");
#endif
#endif
}

template <int N>
__device__ __forceinline__ void wait_asynccnt() {
#if defined(__AMDGCN__)
#if __has_builtin(__builtin_amdgcn_s_wait_asynccnt)
  __builtin_amdgcn_s_wait_asynccnt(N);
#else
  asm volatile("s_wait_asynccnt %0" ::"i"(N) : "memory");
#endif
  asm volatile("" ::: "memory");  // keep LDS reads after the wait
#endif
}

__device__ __forceinline__ void wait_dscnt0() {
#if defined(__AMDGCN__)
  asm volatile("s_wait_dscnt 0" ::: "memory");
#endif
}

// ---------------------------------------------------------------------------
// Issue one group (up to GROUP_ROWS rows, 1KB each) into a wave-private LDS
// buffer. 2 async b128 ops per row (32 lanes * 16B = 512B each); the
// instruction offset applies to BOTH the global and LDS address (ISA §8.2).
// ---------------------------------------------------------------------------
__device__ __forceinline__ void issue_group(const float* __restrict__ feat,
                                            long long row0, long long row_end,
                                            int g, int parity, float* wave_lds,
                                            int lane) {
  float* dst = wave_lds + parity * GROUP_FLOATS;
#pragma unroll
  for (int r = 0; r < GROUP_ROWS; ++r) {
    const long long row = row0 + (long long)g * GROUP_ROWS + r;
    if (row < row_end) {  // wave-uniform predicate -> EXEC stays all-ones
      const float* src = feat + (size_t)row * CHANNELS + lane * 4;
      float* d = dst + r * CHANNELS + lane * 4;
      async_load_b128<0>(src, d);    // channels [0,128)  for this lane slice
      async_load_b128<512>(src, d);  // channels [128,256)
    }
  }
}

__device__ __forceinline__ void flush_acc(float* __restrict__ out, int seg,
                                          int lane, float4& a0, float4& a1) {
  float* p = out + (size_t)seg * CHANNELS + lane * 4;
  atomicAdd(p + 0, a0.x);
  atomicAdd(p + 1, a0.y);
  atomicAdd(p + 2, a0.z);
  atomicAdd(p + 3, a0.w);
  float* q = p + 128;
  atomicAdd(q + 0, a1.x);
  atomicAdd(q + 1, a1.y);
  atomicAdd(q + 2, a1.z);
  atomicAdd(q + 3, a1.w);
  a0 = make_float4(0.f, 0.f, 0.f, 0.f);
  a1 = make_float4(0.f, 0.f, 0.f, 0.f);
}

// ---------------------------------------------------------------------------
// Main kernel (placed FIRST so its disassembly is the visible snippet):
// each wave streams ROWS_PER_WAVE contiguous rows through a double-buffered
// async-LDS pipeline; sigmoid-attention weight per row via wave32 shuffle
// reduction; register accumulation per (sorted) segment, atomic flushes at
// segment boundaries.
// ---------------------------------------------------------------------------
__global__ __launch_bounds__(BLOCK_THREADS) void hgp_pool_kernel(
    const float* __restrict__ feat, const int* __restrict__ batch,
    const float* __restrict__ Watt, const float* __restrict__ batt,
    float* __restrict__ out, int n) {
  __shared__ __align__(16) float lds[BLOCK_LDS_FLOATS];

  const int lane = threadIdx.x & (WAVE_SIZE - 1);
  const int wv = threadIdx.x >> 5;
  const long long wgid = (long long)blockIdx.x * WAVES_PER_BLOCK + wv;
  const long long row0 = wgid * ROWS_PER_WAVE;
  if (row0 >= n) return;  // whole-wave exit (uniform)
  const long long row_end = (row0 + ROWS_PER_WAVE < (long long)n)
                                ? row0 + ROWS_PER_WAVE
                                : (long long)n;
  float* wave_lds = lds + wv * WAVE_LDS_FLOATS;

  // Attention weights resident in registers (8 floats / lane) + bias.
  const float4 wv0 = *(const float4*)(Watt + lane * 4);
  const float4 wv1 = *(const float4*)(Watt + 128 + lane * 4);
  const float bias = batt[0];

  float4 a0 = make_float4(0.f, 0.f, 0.f, 0.f);
  float4 a1 = make_float4(0.f, 0.f, 0.f, 0.f);
  int curSeg = batch[row0];

  const int nGroups = (int)((row_end - row0 + GROUP_ROWS - 1) / GROUP_ROWS);
  issue_group(feat, row0, row_end, 0, 0, wave_lds, lane);

  for (int g = 0; g < nGroups; ++g) {
    const int parity = g & 1;

    // Prefetch group g+1 into the other buffer (its reads from one iteration
    // ago are drained first: async writes are unordered vs DS reads).
    if (g + 1 < nGroups) {
      wait_dscnt0();
      issue_group(feat, row0, row_end, g + 1, parity ^ 1, wave_lds, lane);
    }
    // Group g complete when only group g+1's (full) 2*GROUP_ROWS loads remain.
    if ((g + 1 < nGroups) &&
        (row0 + (long long)(g + 2) * GROUP_ROWS <= row_end))
      wait_asynccnt<2 * GROUP_ROWS>();
    else
      wait_asynccnt<0>();

    const float* buf = wave_lds + parity * GROUP_FLOATS;
    long long remaining = row_end - (row0 + (long long)g * GROUP_ROWS);
    const int rmax = (int)((remaining < GROUP_ROWS) ? remaining : GROUP_ROWS);
    for (int r = 0; r < rmax; ++r) {
      const long long row = row0 + (long long)g * GROUP_ROWS + r;
      const int seg = batch[row];  // sorted -> uniform, monotone
      if (seg != curSeg) {
        flush_acc(out, curSeg, lane, a0, a1);
        curSeg = seg;
      }
      const float4 f0 = *(const float4*)(buf + r * CHANNELS + lane * 4);
      const float4 f1 = *(const float4*)(buf + r * CHANNELS + 128 + lane * 4);
      // per-lane partial dot (8 channels), then wave32 butterfly reduction
      float s = f0.x * wv0.x + f0.y * wv0.y + f0.z * wv0.z + f0.w * wv0.w +
                f1.x * wv1.x + f1.y * wv1.y + f1.z * wv1.z + f1.w * wv1.w;
      s += __shfl_xor(s, 16);
      s += __shfl_xor(s, 8);
      s += __shfl_xor(s, 4);
      s += __shfl_xor(s, 2);
      s += __shfl_xor(s, 1);
      const float wgt = 1.0f / (1.0f + __expf(-(s + bias)));
      a0.x += wgt * f0.x;
      a0.y += wgt * f0.y;
      a0.z += wgt * f0.z;
      a0.w += wgt * f0.w;
      a1.x += wgt * f1.x;
      a1.y += wgt * f1.y;
      a1.z += wgt * f1.z;
      a1.w += wgt * f1.w;
    }
  }
  flush_acc(out, curSeg, lane, a0, a1);
}

// ---------------------------------------------------------------------------
// out[] = 0  (d_out is poisoned by the harness; we accumulate with atomics)
// ---------------------------------------------------------------------------
__global__ void hgp_zero_kernel(float* __restrict__ out, int n4) {
  const int i = blockIdx.x * blockDim.x + threadIdx.x;
  if (i < n4) ((float4*)out)[i] = make_float4(0.f, 0.f, 0.f, 0.f);
}

// ---------------------------------------------------------------------------
extern "C" void kernel_launch(void* const* d_in, const int* in_sizes, int n_in,
                              void* d_out, int out_size, void* d_ws,
                              size_t ws_size, hipStream_t stream) {
  const float* feat = (const float*)d_in[0];   // [N, 256] f32
  const int* batch = (const int*)d_in[1];      // [N] int (sorted)
  const float* Watt = (const float*)d_in[2];   // [256] f32
  const float* batt = (const float*)d_in[3];   // [1] f32
  float* out = (float*)d_out;                  // [2048, 256] f32

  const int n = in_sizes[0] / CHANNELS;

  const int n4 = out_size / 4;
  hgp_zero_kernel<<<(n4 + 255) / 256, 256, 0, stream>>>(out, n4);

  const int waves = (n + ROWS_PER_WAVE - 1) / ROWS_PER_WAVE;
  const int blocks = (waves + WAVES_PER_BLOCK - 1) / WAVES_PER_BLOCK;
  hgp_pool_kernel<<<blocks, BLOCK_THREADS, 0, stream>>>(feat, batch, Watt,
                                                        batt, out, n);
}